// stgp_19499151524250
// MI455X (gfx1250) — compile-verified
//
#include <hip/hip_runtime.h>
#include <math.h>

// ---------------- problem dims ----------------
#define NS 64      // train spatial sites
#define NT 100     // train time points
#define MS 64      // test spatial sites
#define MT 100     // test time points
#define SP 64      // padded spatial dim (multiple of 16)
#define TP 112     // padded time dim (multiple of 16)
#define JITTER 0.001f

typedef __attribute__((ext_vector_type(2))) float v2f;
typedef __attribute__((ext_vector_type(8))) float v8f;

// ---------------- workspace layout (floats) ----------------
constexpr int F_KS  = 0;                // spatial kernel   64x64
constexpr int F_VS  = F_KS  + SP*SP;    // spatial evecs    64x64
constexpr int F_WS  = F_VS  + SP*SP;    // spatial evals    64
constexpr int F_KT  = F_WS  + SP;       // temporal kernel  112x112 (active 100x100)
constexpr int F_VT  = F_KT  + TP*TP;    // temporal evecs   112x112
constexpr int F_WT  = F_VT  + TP*TP;    // temporal evals   112
constexpr int F_Z   = F_WT  + TP;       // calibrated data  64x112
constexpr int F_KSS = F_Z   + SP*TP;    // test spatial     64x64
constexpr int F_KTT = F_KSS + SP*SP;    // test temporal    112x112
constexpr int F_T1  = F_KTT + TP*TP;    // VS^T Z           64x112
constexpr int F_U   = F_T1  + SP*TP;    // T1 VT (then ⊙invD) 64x112
constexpr int F_T2  = F_U   + SP*TP;    // VS W             64x112
constexpr int F_AL  = F_T2  + SP*TP;    // Alpha            64x112
constexpr int F_BS  = F_AL  + SP*TP;    // Ks VS (then ^2)  64x64
constexpr int F_BT  = F_BS  + SP*SP;    // Kt VT (then ^2)  112x112
constexpr int F_ID  = F_BT  + TP*TP;    // invD             64x112
constexpr int F_T3  = F_ID  + SP*TP;    // Ks Alpha         64x112
constexpr int F_C2  = F_T3  + SP*TP;    // BS2 invD         64x112
constexpr int F_YP  = F_C2  + SP*TP;    // yPred padded     64x112
constexpr int F_YV  = F_YP  + SP*TP;    // quad form padded 64x112
constexpr int F_TOT = F_YV  + SP*TP;    // ~138k floats (~554 KB)

// ---------------- utility kernels ----------------
__global__ void zero_ws_k(float* w, int n) {
    int i = blockIdx.x * blockDim.x + threadIdx.x;
    if (i < n) w[i] = 0.0f;
}

__global__ void square_k(float* x, int n) {
    int i = blockIdx.x * blockDim.x + threadIdx.x;
    if (i < n) { float v = x[i]; x[i] = v * v; }
}

// build training-side matrices: spatial kernel, temporal kernel, calibrated data
__global__ void build_train_k(const float* space, const float* timec,
                              const float* stData, const float* log_ll,
                              const float* log_el, const float* log_tl,
                              const float* bias, const float* gain,
                              float* Kspace, float* Ktemp, float* Zcal) {
    float ll = expf(log_ll[0]);
    float el = expf(log_el[0]);
    float tl = expf(log_tl[0]);
    int i = blockIdx.x * blockDim.x + threadIdx.x;
    if (i < NS * NS) {
        int r = i / NS, c = i % NS;
        float dx = (space[r*3+0] - space[c*3+0]) / ll;
        float dy = (space[r*3+1] - space[c*3+1]) / ll;
        float dz = (space[r*3+2] - space[c*3+2]) / el;
        float v = expf(-(dx*dx + dy*dy)) * expf(-(dz*dz));
        if (r == c) v += JITTER;
        Kspace[r*SP + c] = v;
    }
    if (i < NT * NT) {
        int r = i / NT, c = i % NT;
        float dt = (timec[r] - timec[c]) / tl;
        float v = expf(-dt*dt);
        if (r == c) v += JITTER;
        Ktemp[r*TP + c] = v;
    }
    if (i < NS * NT) {
        int s = i / NT, t = i % NT;
        Zcal[s*TP + t] = stData[s*NT + t] * gain[0] + bias[s];
    }
}

// build test-side cross kernels
__global__ void build_test_k(const float* space, const float* timec,
                             const float* tspace, const float* ttime,
                             const float* log_ll, const float* log_el,
                             const float* log_tl,
                             float* Ks, float* Kt) {
    float ll = expf(log_ll[0]);
    float el = expf(log_el[0]);
    float tl = expf(log_tl[0]);
    int i = blockIdx.x * blockDim.x + threadIdx.x;
    if (i < MS * NS) {
        int r = i / NS, c = i % NS;
        float dx = (tspace[r*3+0] - space[c*3+0]) / ll;
        float dy = (tspace[r*3+1] - space[c*3+1]) / ll;
        float dz = (tspace[r*3+2] - space[c*3+2]) / el;
        Ks[r*SP + c] = expf(-(dx*dx + dy*dy)) * expf(-(dz*dz));
    }
    if (i < MT * NT) {
        int r = i / NT, c = i % NT;
        float dt = (ttime[r] - timec[c]) / tl;
        Kt[r*TP + c] = expf(-dt*dt);
    }
}

// cyclic two-sided Jacobi symmetric eigensolver, one workgroup, in-place on A.
// V accumulates eigenvectors (columns); w receives eigenvalues (unsorted — the
// downstream V f(D) V^T reconstruction is order-invariant).
__global__ void jacobi_eigh_k(float* A, float* V, float* w,
                              int n, int ld, int sweeps) {
    const int tid = threadIdx.x, nth = blockDim.x;
    __shared__ float cs[2];
    for (int i = tid; i < n * n; i += nth) {
        int r = i / n, c = i % n;
        V[r*ld + c] = (r == c) ? 1.0f : 0.0f;
    }
    __syncthreads();
    for (int sw = 0; sw < sweeps; ++sw) {
        for (int p = 0; p < n - 1; ++p) {
            for (int q = p + 1; q < n; ++q) {
                if (tid == 0) {
                    float apq = A[p*ld + q];
                    float app = A[p*ld + p], aqq = A[q*ld + q];
                    float c = 1.0f, s = 0.0f;
                    if (fabsf(apq) > 1e-30f) {
                        float tau = (aqq - app) / (2.0f * apq);
                        float t = ((tau >= 0.0f) ? 1.0f : -1.0f) /
                                  (fabsf(tau) + sqrtf(1.0f + tau*tau));
                        c = 1.0f / sqrtf(1.0f + t*t);
                        s = t * c;
                    }
                    cs[0] = c; cs[1] = s;
                }
                __syncthreads();
                float c = cs[0], s = cs[1];
                if (s != 0.0f) {   // uniform branch (shared value)
                    // A <- G^T A  (rows p,q)
                    for (int k = tid; k < n; k += nth) {
                        float akp = A[p*ld + k], akq = A[q*ld + k];
                        A[p*ld + k] = c*akp - s*akq;
                        A[q*ld + k] = s*akp + c*akq;
                    }
                    __syncthreads();
                    // A <- A G (cols p,q); V <- V G
                    for (int k = tid; k < n; k += nth) {
                        float akp = A[k*ld + p], akq = A[k*ld + q];
                        A[k*ld + p] = c*akp - s*akq;
                        A[k*ld + q] = s*akp + c*akq;
                        float vkp = V[k*ld + p], vkq = V[k*ld + q];
                        V[k*ld + p] = c*vkp - s*vkq;
                        V[k*ld + q] = s*vkp + c*vkq;
                    }
                }
                __syncthreads();
            }
        }
    }
    for (int i = tid; i < n; i += nth) w[i] = A[i*ld + i];
}

// invD[s,t] = 1/(wS[s]*wT[t] + sigma_n^2) for active t, 0 in padding;
// also scales U in place: W = U ⊙ invD
__global__ void make_invd_k(const float* wS, const float* wT,
                            const float* log_noise, float* invD, float* Um) {
    float nv = expf(log_noise[0]);
    int i = blockIdx.x * blockDim.x + threadIdx.x;
    if (i < NS * TP) {
        int t = i % TP;
        float d = (t < NT) ? 1.0f / (wS[i / TP] * wT[t] + nv) : 0.0f;
        invD[i] = d;
        Um[i] *= d;
    }
}

// ---------------- FP32 WMMA GEMM (compile-time transpose flags) ----------------
// C[M,N] = opA(A) @ opB(B); one wave per 16x16 tile; K stepped by 4 through
// V_WMMA_F32_16X16X4_F32. M,N multiples of 16; K multiple of 16; zero padding
// in the stored matrices removes all bounds checks, so EXEC stays all-1s
// (WMMA requirement) and the inner loop is branch-free.
//
// A layout (ISA 16x4 f32 table): lanes 0-15 hold K=k0,k0+1 in v.x/v.y;
// lanes 16-31 hold K=k0+2,k0+3. B (4x16) mirrored: lane half selects K pair,
// lane%16 selects N. C/D: VGPR r, lane<16 -> (M=r, N=lane); lane>=16 -> M=r+8.
//
// Contiguous K-pairs (A non-transposed, B transposed) are fetched as v2f
// (-> global_load_b64); strided pairs as two b32 loads. All workspace strides
// and offsets are even, so 8B alignment holds.
template<bool TA, bool TB>
__global__ void wmma_gemm_k(const float* __restrict__ A,
                            const float* __restrict__ B,
                            float* __restrict__ C,
                            int N, int K,
                            int lda, int ldb, int ldc) {
    const int ntilesN = N >> 4;
    const int tile = blockIdx.x;
    const int tm = tile / ntilesN, tn = tile % ntilesN;
    const int lane = threadIdx.x;
    const int half = lane >> 4;       // 0: lanes 0-15, 1: lanes 16-31
    const int l    = lane & 15;

    const int row = tm * 16 + l;      // A logical row for this lane
    const int col = tn * 16 + l;      // B logical col for this lane

    // per-lane base pointer at k = 2*half, and stride per unit k
    const int aStep = TA ? lda : 1;
    const int bStep = TB ? 1 : ldb;
    const float* pA = (TA ? (A + row) : (A + row * lda)) + 2 * half * aStep;
    const float* pB = (TB ? (B + col * ldb) : (B + col)) + 2 * half * bStep;

    v8f acc = {0.f, 0.f, 0.f, 0.f, 0.f, 0.f, 0.f, 0.f};

    #pragma unroll 4
    for (int k0 = 0; k0 < K; k0 += 4) {
        v2f a, b;
        if (TA) { a.x = pA[0]; a.y = pA[aStep]; }
        else    { a = *(const v2f*)pA; }
        if (TB) { b = *(const v2f*)pB; }
        else    { b.x = pB[0]; b.y = pB[bStep]; }
        acc = __builtin_amdgcn_wmma_f32_16x16x4_f32(
            /*neg_a=*/false, a, /*neg_b=*/false, b,
            /*c_mod=*/(short)0, acc, /*reuse_a=*/false, /*reuse_b=*/false);
        pA += 4 * aStep;
        pB += 4 * bStep;
    }

    const int rbase = tm * 16 + half * 8;
    float* pC = C + rbase * ldc + (tn * 16 + l);
    #pragma unroll
    for (int r = 0; r < 8; ++r)
        pC[r * ldc] = acc[r];
}

// yPred and yVar = sigma_s^2 - quadform, depadded into d_out
__global__ void finalize_k(const float* yP, const float* yVm,
                           const float* log_sig, float* out) {
    float sv = expf(log_sig[0]);
    int i = blockIdx.x * blockDim.x + threadIdx.x;
    if (i < MS * MT) {
        int ms = i / MT, mt = i % MT;
        out[i]           = yP[ms*TP + mt];
        out[MS*MT + i]   = sv - yVm[ms*TP + mt];
    }
}

// ---------------- host driver ----------------
extern "C" void kernel_launch(void* const* d_in, const int* in_sizes, int n_in,
                              void* d_out, int out_size, void* d_ws, size_t ws_size,
                              hipStream_t stream) {
    const float* space   = (const float*)d_in[0];   // [64,3]
    const float* timec   = (const float*)d_in[1];   // [100,1]
    const float* stData  = (const float*)d_in[2];   // [64,100]
    const float* tspace  = (const float*)d_in[3];   // [64,3]
    const float* ttime   = (const float*)d_in[4];   // [100,1]
    const float* log_ll  = (const float*)d_in[5];
    const float* log_el  = (const float*)d_in[6];
    const float* log_tl  = (const float*)d_in[7];
    const float* log_nv  = (const float*)d_in[8];
    const float* log_sv  = (const float*)d_in[9];
    const float* bias    = (const float*)d_in[10];  // [64,1]
    const float* gain    = (const float*)d_in[11];  // [1]
    float* out = (float*)d_out;
    float* ws  = (float*)d_ws;

    // compile-time transpose-flag dispatch
    auto gemm_nn = [&](const float* A, const float* B, float* C,
                       int M, int N, int K, int lda, int ldb, int ldc) {
        wmma_gemm_k<false,false><<<dim3((M >> 4) * (N >> 4)), dim3(32), 0, stream>>>(
            A, B, C, N, K, lda, ldb, ldc);
    };
    auto gemm_tn = [&](const float* A, const float* B, float* C,
                       int M, int N, int K, int lda, int ldb, int ldc) {
        wmma_gemm_k<true,false><<<dim3((M >> 4) * (N >> 4)), dim3(32), 0, stream>>>(
            A, B, C, N, K, lda, ldb, ldc);
    };
    auto gemm_nt = [&](const float* A, const float* B, float* C,
                       int M, int N, int K, int lda, int ldb, int ldc) {
        wmma_gemm_k<false,true><<<dim3((M >> 4) * (N >> 4)), dim3(32), 0, stream>>>(
            A, B, C, N, K, lda, ldb, ldc);
    };

    // 0) zero workspace (padding must be zero for padded GEMMs)
    zero_ws_k<<<(F_TOT + 255) / 256, 256, 0, stream>>>(ws, F_TOT);

    // 1) build kernel matrices + calibrated data
    build_train_k<<<(NT*NT + 255) / 256, 256, 0, stream>>>(
        space, timec, stData, log_ll, log_el, log_tl, bias, gain,
        ws + F_KS, ws + F_KT, ws + F_Z);
    build_test_k<<<(MT*NT + 255) / 256, 256, 0, stream>>>(
        space, timec, tspace, ttime, log_ll, log_el, log_tl,
        ws + F_KSS, ws + F_KTT);

    // 2) symmetric eigendecompositions (Jacobi, in-place)
    jacobi_eigh_k<<<1, 128, 0, stream>>>(ws + F_KS, ws + F_VS, ws + F_WS, NS, SP, 16);
    jacobi_eigh_k<<<1, 128, 0, stream>>>(ws + F_KT, ws + F_VT, ws + F_WT, NT, TP, 16);

    // 3) alpha via Kronecker identity: Alpha = VS ((VS^T Z VT) ⊙ invD) VT^T
    gemm_tn(ws + F_VS, ws + F_Z,  ws + F_T1, SP, TP, SP, SP, TP, TP); // T1 = VS^T Z
    gemm_nn(ws + F_T1, ws + F_VT, ws + F_U,  SP, TP, TP, TP, TP, TP); // U  = T1 VT
    make_invd_k<<<(NS*TP + 255) / 256, 256, 0, stream>>>(
        ws + F_WS, ws + F_WT, log_nv, ws + F_ID, ws + F_U);           // W  = U ⊙ invD
    gemm_nn(ws + F_VS, ws + F_U,  ws + F_T2, SP, TP, SP, SP, TP, TP); // T2 = VS W
    gemm_nt(ws + F_T2, ws + F_VT, ws + F_AL, SP, TP, TP, TP, TP, TP); // Alpha = T2 VT^T

    // 4) predictive mean: yPred = Ks Alpha Kt^T
    gemm_nn(ws + F_KSS, ws + F_AL,  ws + F_T3, SP, TP, SP, SP, TP, TP);
    gemm_nt(ws + F_T3,  ws + F_KTT, ws + F_YP, SP, TP, TP, TP, TP, TP);

    // 5) predictive variance: quadform = (Ks VS)^2 invD ((Kt VT)^2)^T
    gemm_nn(ws + F_KSS, ws + F_VS, ws + F_BS, SP, SP, SP, SP, SP, SP); // BS = Ks VS
    gemm_nn(ws + F_KTT, ws + F_VT, ws + F_BT, TP, TP, TP, TP, TP, TP); // BT = Kt VT
    square_k<<<(SP*SP + 255) / 256, 256, 0, stream>>>(ws + F_BS, SP*SP);
    square_k<<<(TP*TP + 255) / 256, 256, 0, stream>>>(ws + F_BT, TP*TP);
    gemm_nn(ws + F_BS, ws + F_ID, ws + F_C2, SP, TP, SP, SP, TP, TP);  // C2 = BS2 invD
    gemm_nt(ws + F_C2, ws + F_BT, ws + F_YV, SP, TP, TP, TP, TP, TP);  // YV = C2 BT2^T

    // 6) depad + variance offset
    finalize_k<<<(MS*MT + 255) / 256, 256, 0, stream>>>(
        ws + F_YP, ws + F_YV, log_sv, out);
}